// SpeMamba_60713657696364
// MI455X (gfx1250) — compile-verified
//
#include <hip/hip_runtime.h>

typedef __attribute__((ext_vector_type(2))) float v2f;
typedef __attribute__((ext_vector_type(8))) float v8f;

#define HW      4096     // 64*64 spatial positions per batch image
#define CH      256
#define THREADS 256

// ---------------------------------------------------------------------------
// Kernel 0: zero the 64-float stats buffer (sum, sumsq per (batch, group))
// ---------------------------------------------------------------------------
__global__ void init_stats(float* __restrict__ stats) {
    if (threadIdx.x < 64) stats[threadIdx.x] = 0.0f;
}

// ---------------------------------------------------------------------------
// Kernel 1: fused Mamba block for 16 sequences per workgroup.
//   Writes y_rec (pre-GroupNorm output, NCHW layout) into d_out and
//   accumulates per-(b,group) sum / sumsq into stats via atomics.
// ---------------------------------------------------------------------------
__global__ __launch_bounds__(256) void mamba_fused(
    const float* __restrict__ xin,
    const float* __restrict__ W_in,     // [32][128]
    const float* __restrict__ conv_w,   // [64][4]
    const float* __restrict__ conv_b,   // [64]
    const float* __restrict__ W_xproj,  // [64][34]
    const float* __restrict__ W_dt,     // [2][64]
    const float* __restrict__ b_dt,     // [64]
    const float* __restrict__ A_log,    // [64][16]
    const float* __restrict__ Dvec,     // [64]
    const float* __restrict__ W_out,    // [64][32]
    float* __restrict__ y_rec,          // = d_out, NCHW
    float* __restrict__ stats)          // [8][4][2]
{
    // Pool A reused across phases: xtile(16x256) -> per-wave xc/y transpose
    // buffers (8 x [16][65]) -> out tile ([256][17])
    __shared__ float poolA[8320];
    __shared__ float xdbAll[4736];      // per-wave [16][37] (dt | B | C)
    __shared__ float Atab[1024];        // A = -exp(A_log), [64][16]
    __shared__ float cwS[256];
    __shared__ float cbS[64];
    __shared__ float wdtS[128];
    __shared__ float bdtS[64];
    __shared__ float DvS[64];
    __shared__ float red[512];

    const int tid    = threadIdx.x;
    const int lane   = tid & 31;
    const int wave   = tid >> 5;          // 0..7
    const int c0     = lane & 15;         // column / N-lane within tile
    const int halfk  = (lane >> 4) << 1;  // K offset: 0 (lanes 0-15) or 2
    const int seqoff = (lane >> 4) << 3;  // row offset 0/8 (which seq of the pair)
    const int lpMine = wave * 2 + (lane >> 4);   // local position 0..15

    const int seqBase = blockIdx.x * 16;         // 16 positions per WG
    const int b       = seqBase >> 12;           // / 4096
    const int pos0    = seqBase & (HW - 1);
    const float* xg   = xin + (size_t)b * (CH * HW) + pos0;

    // ---- stage input tile [pos][ch] + small weights --------------------
    for (int i = tid; i < 4096; i += THREADS) {
        int c = i >> 4, p = i & 15;
        poolA[p * 256 + c] = xg[(size_t)c * HW + p];
    }
    cwS[tid & 255] = conv_w[tid & 255];
    if (tid < 64)  { cbS[tid] = conv_b[tid]; bdtS[tid] = b_dt[tid]; DvS[tid] = Dvec[tid]; }
    if (tid < 128) wdtS[tid] = W_dt[tid];
    for (int i = tid; i < 1024; i += THREADS) Atab[i] = -__expf(A_log[i]);
    __syncthreads();

    // ---- GEMM1: X(16x32) @ W_in(32x128) -> xm | z ----------------------
    // A-fragment row for this lane:
    const int rowA = c0;
    const int pA   = wave * 2 + (rowA >> 3);
    const int tA   = rowA & 7;
    const float* xrow = &poolA[pA * 256 + tA * 32];

    v8f accXM[4], accZ[4];
    #pragma unroll
    for (int i = 0; i < 4; ++i) {
        accXM[i] = (v8f){0.f,0.f,0.f,0.f,0.f,0.f,0.f,0.f};
        accZ[i]  = (v8f){0.f,0.f,0.f,0.f,0.f,0.f,0.f,0.f};
    }
    #pragma unroll
    for (int kk = 0; kk < 8; ++kk) {
        const int kb = kk * 4 + halfk;
        v2f a; a[0] = xrow[kb]; a[1] = xrow[kb + 1];
        #pragma unroll
        for (int nt = 0; nt < 4; ++nt) {
            const float* wp = W_in + kb * 128 + nt * 16 + c0;
            v2f bv; bv[0] = wp[0];  bv[1] = wp[128];
            accXM[nt] = __builtin_amdgcn_wmma_f32_16x16x4_f32(
                false, a, false, bv, (short)0, accXM[nt], false, false);
            v2f bz; bz[0] = wp[64]; bz[1] = wp[64 + 128];
            accZ[nt]  = __builtin_amdgcn_wmma_f32_16x16x4_f32(
                false, a, false, bz, (short)0, accZ[nt], false, false);
        }
    }

    // ---- causal depthwise conv (K=4) + SiLU, all in registers ----------
    // accXM[k][j] = xm[token j of my seq][d = c0 + 16k]
    float xcv[4][8];
    #pragma unroll
    for (int k = 0; k < 4; ++k) {
        const int d = c0 + 16 * k;
        const float w0 = cwS[d*4+0], w1 = cwS[d*4+1], w2 = cwS[d*4+2], w3 = cwS[d*4+3];
        const float cb = cbS[d];
        float p0 = 0.f, p1 = 0.f, p2 = 0.f;
        #pragma unroll
        for (int j = 0; j < 8; ++j) {
            const float cur = accXM[k][j];
            const float s = w0*p0 + w1*p1 + w2*p2 + w3*cur + cb;
            p0 = p1; p1 = p2; p2 = cur;
            xcv[k][j] = s / (1.f + __expf(-s));      // silu
        }
    }

    __syncthreads();                    // xtile reads done -> reuse pool A
    float* xcl = &poolA[wave * 1040];   // per-wave [16][65]
    #pragma unroll
    for (int k = 0; k < 4; ++k)
        #pragma unroll
        for (int j = 0; j < 8; ++j)
            xcl[(seqoff + j) * 65 + c0 + 16 * k] = xcv[k][j];
    __syncthreads();

    // ---- GEMM2: xc(16x64) @ W_xproj(64x34, padded to 48) ---------------
    v8f acc2[3];
    #pragma unroll
    for (int i = 0; i < 3; ++i) acc2[i] = (v8f){0.f,0.f,0.f,0.f,0.f,0.f,0.f,0.f};
    #pragma unroll
    for (int kk = 0; kk < 16; ++kk) {
        const int kb = kk * 4 + halfk;
        v2f a; a[0] = xcl[c0 * 65 + kb]; a[1] = xcl[c0 * 65 + kb + 1];
        #pragma unroll
        for (int nt = 0; nt < 3; ++nt) {
            const int col = nt * 16 + c0;
            v2f bv;
            bv[0] = (col < 34) ? W_xproj[kb * 34 + col]       : 0.f;
            bv[1] = (col < 34) ? W_xproj[(kb + 1) * 34 + col] : 0.f;
            acc2[nt] = __builtin_amdgcn_wmma_f32_16x16x4_f32(
                false, a, false, bv, (short)0, acc2[nt], false, false);
        }
    }
    float* xdb = &xdbAll[wave * 592];   // [16][37]
    #pragma unroll
    for (int nt = 0; nt < 3; ++nt) {
        const int cc = nt * 16 + c0;
        if (cc < 36)
            #pragma unroll
            for (int j = 0; j < 8; ++j)
                xdb[(seqoff + j) * 37 + cc] = acc2[nt][j];
    }
    __syncthreads();

    // ---- per-lane SSM scan: seq = my half, d = c0 + 16k, n = 0..15 -----
    float wd0[4], wd1[4], bd[4], Dd[4];
    #pragma unroll
    for (int k = 0; k < 4; ++k) {
        const int d = c0 + 16 * k;
        wd0[k] = wdtS[d]; wd1[k] = wdtS[64 + d]; bd[k] = bdtS[d]; Dd[k] = DvS[d];
    }
    float h[4][16];
    #pragma unroll
    for (int k = 0; k < 4; ++k)
        #pragma unroll
        for (int n = 0; n < 16; ++n) h[k][n] = 0.f;

    float* ylds = xcl;                  // reuse my wave's pool-A slice
    #pragma unroll
    for (int t = 0; t < 8; ++t) {
        const int row = seqoff + t;
        const float dt0 = xdb[row * 37 + 0];
        const float dt1 = xdb[row * 37 + 1];
        float Bt[16], Ct[16];
        #pragma unroll
        for (int n = 0; n < 16; ++n) {
            Bt[n] = xdb[row * 37 + 2 + n];
            Ct[n] = xdb[row * 37 + 18 + n];
        }
        #pragma unroll
        for (int k = 0; k < 4; ++k) {
            const int d = c0 + 16 * k;
            const float pre   = dt0 * wd0[k] + dt1 * wd1[k] + bd[k];
            const float delta = (pre > 20.f) ? pre : __logf(1.f + __expf(pre));
            const float xct = xcv[k][t];
            const float dx  = delta * xct;
            float y = 0.f;
            #pragma unroll
            for (int n = 0; n < 16; ++n) {
                const float dA = __expf(delta * Atab[d * 16 + n]);
                h[k][n] = dA * h[k][n] + dx * Bt[n];
                y += h[k][n] * Ct[n];
            }
            const float zv = accZ[k][t];
            const float sz = zv / (1.f + __expf(-zv));   // silu(z)
            ylds[row * 65 + d] = (y + xct * Dd[k]) * sz;
        }
    }
    __syncthreads();

    // ---- GEMM3: y(16x64) @ W_out(64x32) --------------------------------
    v8f acc3[2];
    acc3[0] = (v8f){0.f,0.f,0.f,0.f,0.f,0.f,0.f,0.f};
    acc3[1] = (v8f){0.f,0.f,0.f,0.f,0.f,0.f,0.f,0.f};
    #pragma unroll
    for (int kk = 0; kk < 16; ++kk) {
        const int kb = kk * 4 + halfk;
        v2f a; a[0] = ylds[c0 * 65 + kb]; a[1] = ylds[c0 * 65 + kb + 1];
        #pragma unroll
        for (int nt = 0; nt < 2; ++nt) {
            v2f bv;
            bv[0] = W_out[kb * 32 + nt * 16 + c0];
            bv[1] = W_out[(kb + 1) * 32 + nt * 16 + c0];
            acc3[nt] = __builtin_amdgcn_wmma_f32_16x16x4_f32(
                false, a, false, bv, (short)0, acc3[nt], false, false);
        }
    }
    __syncthreads();                    // everyone done reading pool A

    // ---- transpose out tile through LDS: [ch][pos], stride 17 ----------
    float* ot = poolA;
    #pragma unroll
    for (int nt = 0; nt < 2; ++nt)
        #pragma unroll
        for (int j = 0; j < 8; ++j) {
            const int chn = j * 32 + nt * 16 + c0;   // token j -> channels j*32..
            ot[chn * 17 + lpMine] = acc3[nt][j];
        }
    __syncthreads();

    // ---- coalesced global store of y_rec -------------------------------
    const size_t obase = (size_t)b * (CH * HW) + pos0;
    for (int i = tid; i < 4096; i += THREADS) {
        int c = i >> 4, p = i & 15;
        y_rec[obase + (size_t)c * HW + p] = ot[c * 17 + p];
    }

    // ---- partial GroupNorm stats (4 groups of 64 channels) -------------
    {
        float s = 0.f, s2 = 0.f;
        #pragma unroll
        for (int p = 0; p < 16; ++p) {
            const float v = ot[tid * 17 + p];
            s += v; s2 += v * v;
        }
        red[tid] = s; red[256 + tid] = s2;
    }
    __syncthreads();
    if (tid < 8) {
        const int g = tid >> 1, which = tid & 1;
        float acc = 0.f;
        for (int c = g * 64; c < g * 64 + 64; ++c) acc += red[which * 256 + c];
        atomicAdd(&stats[(b * 4 + g) * 2 + which], acc);
    }
}

// ---------------------------------------------------------------------------
// Kernel 2: GroupNorm + SiLU + residual, in place on d_out
// ---------------------------------------------------------------------------
__global__ __launch_bounds__(256) void gn_finalize(
    const float* __restrict__ xin,
    const float* __restrict__ gn_w,
    const float* __restrict__ gn_b,
    const float* __restrict__ stats,
    float* __restrict__ out)
{
    const float inv = 1.0f / 262144.0f;   // 64 channels * 4096 positions
    const size_t total = (size_t)8 * CH * HW;
    for (size_t idx = (size_t)blockIdx.x * blockDim.x + threadIdx.x;
         idx < total; idx += (size_t)gridDim.x * blockDim.x) {
        const int bb = (int)(idx >> 20);
        const int ch = (int)((idx >> 12) & 255);
        const int g  = ch >> 6;
        const int sb = (bb * 4 + g) * 2;
        const float mu  = stats[sb] * inv;
        const float var = stats[sb + 1] * inv - mu * mu;
        const float rs  = rsqrtf(var + 1e-5f);
        const float y   = out[idx];
        const float t   = (y - mu) * rs * gn_w[ch] + gn_b[ch];
        const float si  = t / (1.f + __expf(-t));     // silu
        out[idx] = xin[idx] + si;
    }
}

// ---------------------------------------------------------------------------
extern "C" void kernel_launch(void* const* d_in, const int* in_sizes, int n_in,
                              void* d_out, int out_size, void* d_ws, size_t ws_size,
                              hipStream_t stream) {
    const float* x       = (const float*)d_in[0];
    const float* W_in    = (const float*)d_in[1];
    const float* conv_w  = (const float*)d_in[2];
    const float* conv_b  = (const float*)d_in[3];
    const float* W_xproj = (const float*)d_in[4];
    const float* W_dt    = (const float*)d_in[5];
    const float* b_dt    = (const float*)d_in[6];
    const float* A_log   = (const float*)d_in[7];
    const float* Dvec    = (const float*)d_in[8];
    const float* W_out   = (const float*)d_in[9];
    const float* gn_w    = (const float*)d_in[10];
    const float* gn_b    = (const float*)d_in[11];
    float* out   = (float*)d_out;
    float* stats = (float*)d_ws;

    init_stats<<<1, 64, 0, stream>>>(stats);
    mamba_fused<<<2048, THREADS, 0, stream>>>(
        x, W_in, conv_w, conv_b, W_xproj, W_dt, b_dt, A_log, Dvec, W_out,
        out, stats);
    gn_finalize<<<4096, THREADS, 0, stream>>>(x, gn_w, gn_b, stats, out);
}